// ModelNew_69879117906300
// MI455X (gfx1250) — compile-verified
//
#include <hip/hip_runtime.h>
#include <math.h>

#define BATCH 8
#define IN_SZ 8192
#define HID_SZ 8192
#define OUT_SZ 2048
#define KDIM (IN_SZ + HID_SZ)   // 16384

typedef float v2f __attribute__((ext_vector_type(2)));
typedef float v8f __attribute__((ext_vector_type(8)));

// -----------------------------------------------------------------------------
// Pack concat(x, h) into per-lane A-operand order for V_WMMA_F32_16X16X4_F32.
// aPack layout: [KDIM/4 chunks][32 lanes][2 floats]
//   lane < 16 : holds A[m=lane][k0+0], A[m=lane][k0+1]
//   lane >= 16: holds A[m=lane-16][k0+2], A[m=lane-16][k0+3]
// Rows m = 8..15 are zero padding (batch is 8).
// -----------------------------------------------------------------------------
__global__ void pack_act_kernel(const float* __restrict__ x,
                                const float* __restrict__ h,
                                float* __restrict__ aPack) {
    int tid   = blockIdx.x * blockDim.x + threadIdx.x;  // chunk*32 + lane
    int lane  = tid & 31;
    int chunk = tid >> 5;
    int m     = lane & 15;
    int kb    = (chunk << 2) + ((lane >> 4) << 1);

    float v0 = 0.0f, v1 = 0.0f;
    if (m < BATCH) {
        // kb is even and IN_SZ is even, so kb and kb+1 fall on the same side
        if (kb < IN_SZ) {
            v0 = x[(size_t)m * IN_SZ + kb];
            v1 = x[(size_t)m * IN_SZ + kb + 1];
        } else {
            v0 = h[(size_t)m * HID_SZ + (kb - IN_SZ)];
            v1 = h[(size_t)m * HID_SZ + (kb - IN_SZ) + 1];
        }
    }
    v2f v;
    v.x = v0;
    v.y = v1;
    *(v2f*)(aPack + (size_t)tid * 2) = v;
}

// -----------------------------------------------------------------------------
// Streaming GEMM:  D[m][n] = sum_k A[m][k] * W[n][k]  (+ bias, optional tanh)
//   - one workgroup per 16-wide output tile (N dimension)
//   - 8 waves split the K dimension, LDS tree-reduce at the end
//   - B operand: lane r<16 loads W[n0+r][k..k+1], lanes>=16 load k+2..k+3
//     (contiguous b64 per lane, matches the 4x16 fp32 B layout)
//   - WRITE_HIDDEN: apply tanh and emit in packed A-layout for the next GEMM
// -----------------------------------------------------------------------------
template <int KD, int NTOT, bool WRITE_HIDDEN>
__global__ void wmma_gemm_kernel(const float* __restrict__ W,
                                 const float* __restrict__ apack,
                                 const float* __restrict__ bias,
                                 float* __restrict__ dst) {
    const int lane    = threadIdx.x & 31;
    const int wave    = threadIdx.x >> 5;
    const int n0      = blockIdx.x * 16;
    const int nchunks = KD / 4;          // K-chunks of 4
    const int perwave = nchunks / 8;     // 8 waves per workgroup
    const int c0      = wave * perwave;

    const float* wrow = W + (size_t)(n0 + (lane & 15)) * KD + ((lane >> 4) << 1);
    const float* ap   = apack + ((size_t)c0 * 32 + lane) * 2;

    v8f acc = {};
    for (int c = 0; c < perwave; c += 8) {
        // one cacheline-per-row prefetch, ~1KB ahead in the K stream
        int pf = c + 64;
        if (pf > perwave - 8) pf = perwave - 8;
        __builtin_prefetch(wrow + (size_t)(c0 + pf) * 4, 0, 1);
#pragma unroll
        for (int u = 0; u < 8; ++u) {
            v2f a = *(const v2f*)(ap + (size_t)(c + u) * 64);
            v2f b = *(const v2f*)(wrow + (size_t)(c0 + c + u) * 4);
            acc = __builtin_amdgcn_wmma_f32_16x16x4_f32(
                /*neg_a=*/false, a, /*neg_b=*/false, b,
                /*c_mod=*/(short)0, acc, /*reuse_a=*/false, /*reuse_b=*/false);
        }
    }

    // Reduce the 8 per-wave partial C tiles through LDS.
    __shared__ float red[8 * 256];
#pragma unroll
    for (int j = 0; j < 8; ++j)
        red[wave * 256 + j * 32 + lane] = acc[j];
    __syncthreads();

    const int tid = threadIdx.x;  // 0..255 == one C element each
    float s = 0.0f;
#pragma unroll
    for (int w = 0; w < 8; ++w)
        s += red[w * 256 + tid];

    // C layout: VGPR j, lanes 0-15 -> M=j, lanes 16-31 -> M=j+8; N = lane%16.
    const int j = tid >> 5;
    const int l = tid & 31;
    const int m = j + ((l >> 4) << 3);
    const int n = n0 + (l & 15);

    if (WRITE_HIDDEN) {
        float hv = 0.0f;
        if (m < BATCH) hv = tanhf(s + bias[n]);
        // scatter into packed A-layout (K index = n) for the next GEMM;
        // padding rows (m>=8) written as 0.
        size_t idx = ((size_t)(n >> 2) * 32 + (((n >> 1) & 1) << 4) + m) * 2 + (n & 1);
        dst[idx] = hv;
    } else {
        if (m < BATCH) dst[(size_t)m * NTOT + n] = s + bias[n];
    }
}

extern "C" void kernel_launch(void* const* d_in, const int* in_sizes, int n_in,
                              void* d_out, int out_size, void* d_ws, size_t ws_size,
                              hipStream_t stream) {
    const float* x    = (const float*)d_in[0];  // [8][8192]
    const float* h0   = (const float*)d_in[1];  // [8][8192]
    const float* w1   = (const float*)d_in[2];  // [8192][16384]
    const float* b1   = (const float*)d_in[3];  // [8192]
    const float* w2   = (const float*)d_in[4];  // [2048][8192]
    const float* b2   = (const float*)d_in[5];  // [2048]
    float*       out  = (float*)d_out;          // [8][2048]

    float* aPack = (float*)d_ws;                          // (16384/4)*64 floats = 1 MB
    float* hPack = aPack + (size_t)(KDIM / 4) * 64;       // (8192/4)*64 floats = 512 KB

    // 1) pack concat(x, h) into WMMA A-operand order
    pack_act_kernel<<<(KDIM / 4) * 32 / 256, 256, 0, stream>>>(x, h0, aPack);

    // 2) hidden = tanh(A @ W1^T + b1), emitted in packed A-layout
    wmma_gemm_kernel<KDIM, HID_SZ, true>
        <<<HID_SZ / 16, 256, 0, stream>>>(w1, aPack, b1, hPack);

    // 3) out = hidden @ W2^T + b2
    wmma_gemm_kernel<HID_SZ, OUT_SZ, false>
        <<<OUT_SZ / 16, 256, 0, stream>>>(w2, hPack, b2, out);
}